// DynamicDistMatchingLoss_32478542693116
// MI455X (gfx1250) — compile-verified
//
#include <hip/hip_runtime.h>
#include <hip/hip_bf16.h>
#include <math.h>

// ---------------------------------------------------------------------------
// DynamicDistMatchingLoss on gfx1250 (MI455X).
// Heavy part: for each of K=4 classes (N=1M samples, d=8) evaluate the
// Mahalanobis form under all 4 components. y = W_j * x - (W_j * mu_j) is an
// 8x8 triangular matvec -> mapped onto V_WMMA_F32_16X16X4_F32:
//   A (16x4 f32): rows 0..7 = W_{j0}, rows 8..15 = W_{j1}, K-chunked by 4
//   B (4x16 f32): 16 samples, 4 of the 8 feature rows per chunk
//   C (16x16 f32): bias -W_j*mu_j broadcast along N
// Two chained WMMAs per component-pair cover d=8; two pairs cover 4 comps.
// ---------------------------------------------------------------------------

typedef float v2f __attribute__((ext_vector_type(2)));
typedef float v8f __attribute__((ext_vector_type(8)));

#define DDM_K 4
#define DDM_D 8

// ws layout (floats):
//   [0   .. 255] : W[j][r][c]  (L^-1, full 8x8, upper zero), j-major
//   [256 .. 287] : v[j][r] = (W_j * mu_j)[r]
//   [288 .. 291] : cst[j] = 0.5*d*log(2*pi) - sum(log diag L_j)

__global__ void ddml_zero_out(float* out) { out[0] = 0.0f; }

__global__ void ddml_precompute(const float* __restrict__ means,
                                const float* __restrict__ covs,
                                float* __restrict__ ws) {
  if (threadIdx.x != 0 || blockIdx.x != 0) return;
  const float LOG_2PI = 1.8378770664093453f;
  for (int j = 0; j < DDM_K; ++j) {
    float L[DDM_D][DDM_D];
    for (int r = 0; r < DDM_D; ++r)
      for (int c = 0; c < DDM_D; ++c)
        L[r][c] = covs[j * DDM_D * DDM_D + r * DDM_D + c];
    // In-place Cholesky (lower).
    for (int r = 0; r < DDM_D; ++r) {
      for (int c = 0; c <= r; ++c) {
        float s = L[r][c];
        for (int k = 0; k < c; ++k) s -= L[r][k] * L[c][k];
        L[r][c] = (r == c) ? sqrtf(s) : (s / L[c][c]);
      }
      for (int c = r + 1; c < DDM_D; ++c) L[r][c] = 0.0f;
    }
    float hld = 0.0f;
    for (int r = 0; r < DDM_D; ++r) hld += logf(L[r][r]);
    // W = L^{-1} via forward substitution on identity.
    float W[DDM_D][DDM_D];
    for (int r = 0; r < DDM_D; ++r)
      for (int c = 0; c < DDM_D; ++c) W[r][c] = 0.0f;
    for (int c = 0; c < DDM_D; ++c) {
      for (int r = c; r < DDM_D; ++r) {
        float s = (r == c) ? 1.0f : 0.0f;
        for (int k = c; k < r; ++k) s -= L[r][k] * W[k][c];
        W[r][c] = s / L[r][r];
      }
    }
    for (int r = 0; r < DDM_D; ++r)
      for (int c = 0; c < DDM_D; ++c)
        ws[j * 64 + r * 8 + c] = W[r][c];
    for (int r = 0; r < DDM_D; ++r) {
      float v = 0.0f;
      for (int c = 0; c <= r; ++c) v += W[r][c] * means[j * DDM_D + c];
      ws[256 + j * 8 + r] = v;
    }
    ws[288 + j] = 0.5f * (float)DDM_D * LOG_2PI - hld;
  }
}

// Exchange a float with the partner lane (lane ^ 16) using DS swizzle
// (group-of-32, xor_mask=0x10, and_mask=0x1f -> offset 0x401F).
__device__ __forceinline__ float ddml_xor16(float x) {
  int xi = __builtin_amdgcn_ds_swizzle(__float_as_int(x), 0x401F);
  return __int_as_float(xi);
}

__global__ void __launch_bounds__(256)
ddml_main(const float* __restrict__ pred,   // (K, d, N)
          const int* __restrict__ indices,  // (K)
          const float* __restrict__ ws,
          float* __restrict__ out,
          int N) {
  const int i = blockIdx.y;           // class / sample-set index
  const int ci = indices[i];
  if (ci == 0) return;                // do_bg == False
  int cnt = 0;
  for (int k = 0; k < DDM_K; ++k) cnt += (indices[k] != 0) ? 1 : 0;
  const float scale = 1.0f / ((float)N * (float)cnt);  // class_weight = 1

  const int lane = threadIdx.x & 31;
  const int wave = threadIdx.x >> 5;
  const int half = lane >> 4;         // 0: lanes 0-15, 1: lanes 16-31
  const int sl   = lane & 15;         // sample slot within tile / A row (M)
  const int k0   = half * 2;          // A/B K-column base for this wave half

  // --- A matrices (loop invariant). Pair1 = comps (0,1), Pair2 = (2,3). ---
  const int compP1 = (sl < 8) ? 0 : 1;
  const int r1     = sl & 7;
  const float* W1 = ws + compP1 * 64 + r1 * 8;
  const float* W2 = ws + (compP1 + 2) * 64 + r1 * 8;
  v2f a1c1; a1c1.x = W1[k0];     a1c1.y = W1[k0 + 1];
  v2f a1c2; a1c2.x = W1[k0 + 4]; a1c2.y = W1[k0 + 5];
  v2f a2c1; a2c1.x = W2[k0];     a2c1.y = W2[k0 + 1];
  v2f a2c2; a2c2.x = W2[k0 + 4]; a2c2.y = W2[k0 + 5];

  // --- C bias: this lane's D rows are M = v + 8*half -> comp = half (pair1),
  //     comp = 2+half (pair2). C[r][n] = -(W_j mu_j)[r] for all n. ---
  const float* vb1 = ws + 256 + half * 8;
  const float* vb2 = ws + 256 + (2 + half) * 8;
  v8f c1, c2;
#pragma unroll
  for (int v = 0; v < 8; ++v) { c1[v] = -vb1[v]; c2[v] = -vb2[v]; }

  const int   jA = half, jB = 2 + half;          // local component positions
  const float cstA = ws[288 + jA];
  const float cstB = ws[288 + jB];
  const int   idxA = indices[jA], idxB = indices[jB];

  const float* base = pred + (size_t)i * DDM_D * (size_t)N;

  float acc = 0.0f;
  const int tiles = (N + 15) >> 4;
  const int wavesPerBlock = blockDim.x >> 5;
  const int waveStride = gridDim.x * wavesPerBlock;
  for (int t = blockIdx.x * wavesPerBlock + wave; t < tiles; t += waveStride) {
    const int n  = t * 16 + sl;
    const int nc = (n < N) ? n : (N - 1);
    // B (4x16): row K = feature c; lanes 0-15 carry K={k0,k0+1}, lanes 16-31
    // carry K={k0+2,k0+3} per the 32-bit A/B lane split. Coalesced 64B runs.
    v2f b1, b2;
    b1.x = base[(size_t)(k0)     * N + nc];
    b1.y = base[(size_t)(k0 + 1) * N + nc];
    b2.x = base[(size_t)(k0 + 4) * N + nc];
    b2.y = base[(size_t)(k0 + 5) * N + nc];

    v8f d1 = __builtin_amdgcn_wmma_f32_16x16x4_f32(false, a1c1, false, b1,
                                                   (short)0, c1, false, false);
    d1     = __builtin_amdgcn_wmma_f32_16x16x4_f32(false, a1c2, false, b2,
                                                   (short)0, d1, false, false);
    v8f d2 = __builtin_amdgcn_wmma_f32_16x16x4_f32(false, a2c1, false, b1,
                                                   (short)0, c2, false, false);
    d2     = __builtin_amdgcn_wmma_f32_16x16x4_f32(false, a2c2, false, b2,
                                                   (short)0, d2, false, false);

    // Mahalanobis forms for this lane's two components at sample `sl`.
    float mA = 0.0f, mB = 0.0f;
#pragma unroll
    for (int v = 0; v < 8; ++v) {
      mA = fmaf(d1[v], d1[v], mA);
      mB = fmaf(d2[v], d2[v], mB);
    }
    const float lpA = 0.5f * mA + cstA;   // lp as written in the reference
    const float lpB = 0.5f * mB + cstB;

    float eloc = ((idxA != ci) ? expf(lpA) : 0.0f) +
                 ((idxB != ci) ? expf(lpB) : 0.0f);
    float tloc = ((jA == ci) ? lpA : 0.0f) + ((jB == ci) ? lpB : 0.0f);
    // Assemble all four components for the sample across wave halves.
    const float eoth = ddml_xor16(eloc);
    const float toth = ddml_xor16(tloc);
    const float tot = eloc + eoth + 1e-8f;
    const float tgt = tloc + toth;
    const float contrib = logf(tot + expf(tgt)) - tgt;
    if (n < N) acc += 0.5f * contrib;     // both halves computed it -> x0.5
  }

  __shared__ float red[256];
  red[threadIdx.x] = acc;
  __syncthreads();
  for (int s = blockDim.x >> 1; s > 0; s >>= 1) {
    if ((int)threadIdx.x < s) red[threadIdx.x] += red[threadIdx.x + s];
    __syncthreads();
  }
  if (threadIdx.x == 0) atomicAdd(out, red[0] * scale);
}

extern "C" void kernel_launch(void* const* d_in, const int* in_sizes, int n_in,
                              void* d_out, int out_size, void* d_ws, size_t ws_size,
                              hipStream_t stream) {
  const float* pred    = (const float*)d_in[0];  // (K, d, N) f32
  const float* means   = (const float*)d_in[1];  // (K, d)    f32
  const float* covs    = (const float*)d_in[2];  // (K, d, d) f32
  const int*   indices = (const int*)d_in[3];    // (K)       i32
  float* out = (float*)d_out;                    // scalar f32
  float* ws  = (float*)d_ws;

  const int K = in_sizes[3];                 // 4
  const int d = in_sizes[1] / K;             // 8
  const int N = in_sizes[0] / (K * d);       // 1048576
  (void)n_in; (void)out_size; (void)ws_size; (void)K; (void)d;

  ddml_zero_out<<<1, 1, 0, stream>>>(out);
  ddml_precompute<<<1, 1, 0, stream>>>(means, covs, ws);

  dim3 grid(512, DDM_K);
  ddml_main<<<grid, dim3(256), 0, stream>>>(pred, indices, ws, out, N);
}